// ParametricKacLayer_72688026517802
// MI455X (gfx1250) — compile-verified
//
#include <hip/hip_runtime.h>
#include <math.h>

typedef float v2f __attribute__((ext_vector_type(2)));
typedef float v8f __attribute__((ext_vector_type(8)));
typedef unsigned int u32x4 __attribute__((ext_vector_type(4)));
typedef int i32x4 __attribute__((ext_vector_type(4)));
typedef int i32x8 __attribute__((ext_vector_type(8)));

#define TILE_M 128
#define TILE_N 64
#define TILE_K 32
#define LDSS   36   // 32 dwords data + 4 dwords TDM pad: even (aligned b64 frags),
                    // 18*d = -1 (mod 32) unsolvable -> both lane halves conflict-free

// ---------------------------------------------------------------------------
// Phase 0: pack per-step {i, j, cos, sin} into one 16B record.
// ---------------------------------------------------------------------------
__global__ __launch_bounds__(256) void kac_build_table(
    const float* __restrict__ angles, const int* __restrict__ pi,
    const int* __restrict__ pj, const int* __restrict__ aidx,
    float4* __restrict__ table, int ns)
{
    int s = blockIdx.x * blockDim.x + threadIdx.x;
    if (s >= ns) return;
    float th = angles[aidx[s]];
    float sn, cs;
    sincosf(th, &sn, &cs);
    float4 e;
    e.x = __int_as_float(pi[s]);
    e.y = __int_as_float(pj[s]);
    e.z = cs;
    e.w = sn;
    table[s] = e;
}

// ---------------------------------------------------------------------------
// Phase 1a: Q = I, column-major  Q[n*dim + k].
// ---------------------------------------------------------------------------
__global__ __launch_bounds__(256) void kac_init_q(float* __restrict__ q, int dim)
{
    size_t total  = (size_t)dim * (size_t)dim;
    size_t stride = (size_t)gridDim.x * blockDim.x;
    for (size_t idx = (size_t)blockIdx.x * blockDim.x + threadIdx.x; idx < total; idx += stride)
        q[idx] = ((idx / (size_t)dim) == (idx % (size_t)dim)) ? 1.0f : 0.0f;
}

// ---------------------------------------------------------------------------
// Phase 1b: apply all Givens rotations to Q's columns. Thread t owns row t:
// rows are independent under column rotations, so the 16K-step chain needs no
// synchronization. Column-major -> lanes hit consecutive addresses; the step
// record is a wave-uniform s_load_b128; Q (64MB) stays resident in L2.
// ---------------------------------------------------------------------------
__global__ __launch_bounds__(256) void kac_rotate_q(
    float* __restrict__ q, const float4* __restrict__ table, int dim, int ns)
{
    int t = blockIdx.x * blockDim.x + threadIdx.x;
    if (t >= dim) return;
    for (int s = 0; s < ns; ++s) {
        float4 e = table[s];
        int   i  = __float_as_int(e.x);
        int   j  = __float_as_int(e.y);
        float cs = e.z, sn = e.w;
        size_t oi = (size_t)i * dim + t;
        size_t oj = (size_t)j * dim + t;
        float qi = q[oi];
        float qj = q[oj];
        q[oi] = fmaf(cs, qi, sn * qj);
        q[oj] = fmaf(-sn, qi, cs * qj);
    }
}

// ---------------------------------------------------------------------------
// Tensor Data Mover: load a 2D tile (tile_d0 contiguous elems x tile_d1 rows,
// row stride = row_stride elems, f32) into LDS at lds_addr, padding the LDS
// destination by 4 dwords after every 32 dwords (-> stride LDSS=36).
// D# layout per CDNA5 ISA ch.8 (group0 4dw, group1 8dw); groups for dims 3+/
// gather are unused (2D tile) and passed as zeros. 6-arg clang-23 builtin.
// ---------------------------------------------------------------------------
__device__ __forceinline__ void tdm_load_tile_f32(
    unsigned lds_addr, const float* gptr, int tile_d0, int tile_d1, long row_stride)
{
    unsigned long long ga = (unsigned long long)(uintptr_t)gptr;
    u32x4 g0;
    g0[0] = 1u;                                       // count=1, no gather
    g0[1] = lds_addr;                                 // LDS byte address
    g0[2] = (unsigned)(ga & 0xFFFFFFFFu);             // global_addr[31:0]
    g0[3] = (unsigned)((ga >> 32) & 0x01FFFFFFu)      // global_addr[56:32]
          | 0x80000000u;                              // type=2 (bits 127:126)

    unsigned td0 = (unsigned)tile_d0;                 // tensor dims = tile (in-bounds)
    unsigned td1 = (unsigned)tile_d1;
    unsigned long long s0 = (unsigned long long)row_stride;
    i32x8 g1;
    g1[0] = (int)((2u << 16)                          // data_size = 4B
                | (1u << 20)                          // pad_enable
                | (4u << 22)                          // pad_interval: every 32 dwords
                | (3u << 25));                        // pad_amount: 4 dwords
    g1[1] = (int)((td0 & 0xFFFFu) << 16);             // tensor_dim0[15:0] (atomic addr=0)
    g1[2] = (int)((td0 >> 16) | ((td1 & 0xFFFFu) << 16)); // tensor_dim0[31:16] | tensor_dim1[15:0]
    g1[3] = (int)((td1 >> 16) | ((unsigned)tile_d0 << 16)); // tensor_dim1[31:16] | tile_dim0
    g1[4] = (int)((unsigned)tile_d1 & 0xFFFFu);       // tile_dim1 (tile_dim2 = 0)
    g1[5] = (int)(s0 & 0xFFFFFFFFu);                  // tensor_dim0_stride[31:0]
    g1[6] = (int)((s0 >> 32) & 0xFFFFu);              // tensor_dim0_stride[47:32]
    g1[7] = 0;
    i32x4 z4 = {0, 0, 0, 0};
    i32x8 z8 = {0, 0, 0, 0, 0, 0, 0, 0};
    __builtin_amdgcn_tensor_load_to_lds(g0, g1, z4, z4, z8, 0);
}

// ---------------------------------------------------------------------------
// Phase 2: y = x @ Q, full-f32 GEMM via V_WMMA_F32_16X16X4_F32.
// 128x64 tile / 256 threads (8 waves, 4x2 -> 32x32 per wave).
// A and B both staged by the TDM into identical [row][k] stride-36 layouts;
// double-buffered: wave 0 prefetches chunk k+1 while all waves do WMMAs.
// ---------------------------------------------------------------------------
__global__ __launch_bounds__(256) void kac_gemm_wmma(
    const float* __restrict__ X,   // [M x K] row-major
    const float* __restrict__ Qc,  // [K x N] stored column-major: Qc[n*K + k]
    float* __restrict__ Y,         // [M x N] row-major
    int M, int N, int K)
{
    __shared__ float As[2][TILE_M * LDSS];
    __shared__ float Bs[2][TILE_N * LDSS];

    const int tid      = threadIdx.x;
    const int lane     = tid & 31;
    const int wave     = tid >> 5;
    const int laneHalf = lane >> 4;   // 0 -> K{0,1}, 1 -> K{2,3}
    const int laneMod  = lane & 15;

    const int mTile = blockIdx.y * TILE_M;
    const int nTile = blockIdx.x * TILE_N;
    const int wM    = (wave & 3) * 32;   // 4 waves along M
    const int wN    = (wave >> 2) * 32;  // 2 waves along N

    const unsigned ldsA0 = (unsigned)(uintptr_t)&As[0][0];
    const unsigned ldsA1 = (unsigned)(uintptr_t)&As[1][0];
    const unsigned ldsB0 = (unsigned)(uintptr_t)&Bs[0][0];
    const unsigned ldsB1 = (unsigned)(uintptr_t)&Bs[1][0];

    v8f c[2][2];
#pragma unroll
    for (int mt = 0; mt < 2; ++mt)
#pragma unroll
        for (int nt = 0; nt < 2; ++nt) {
            v8f z = {0.f, 0.f, 0.f, 0.f, 0.f, 0.f, 0.f, 0.f};
            c[mt][nt] = z;
        }

    // Prologue: TDM-stage chunk 0 into buffer 0.
    if (wave == 0) {
        tdm_load_tile_f32(ldsA0, X  + (size_t)mTile * K, TILE_K, TILE_M, K);
        tdm_load_tile_f32(ldsB0, Qc + (size_t)nTile * K, TILE_K, TILE_N, K);
        __builtin_amdgcn_s_wait_tensorcnt(0);
    }
    __syncthreads();

    int buf = 0;
    for (int k0 = 0; k0 < K; k0 += TILE_K) {
        // Prefetch next chunk into the other buffer while this one is consumed.
        if (wave == 0 && (k0 + TILE_K) < K) {
            unsigned la = buf ? ldsA0 : ldsA1;
            unsigned lb = buf ? ldsB0 : ldsB1;
            tdm_load_tile_f32(la, X  + (size_t)mTile * K + (k0 + TILE_K), TILE_K, TILE_M, K);
            tdm_load_tile_f32(lb, Qc + (size_t)nTile * K + (k0 + TILE_K), TILE_K, TILE_N, K);
        }

        const float* as = &As[buf][0];
        const float* bs = &Bs[buf][0];
#pragma unroll
        for (int ks = 0; ks < TILE_K; ks += 4) {
            v2f a[2], b[2];
#pragma unroll
            for (int mt = 0; mt < 2; ++mt) {
                int m = wM + mt * 16 + laneMod;
                a[mt] = *(const v2f*)(as + m * LDSS + ks + laneHalf * 2);
            }
#pragma unroll
            for (int nt = 0; nt < 2; ++nt) {
                int n = wN + nt * 16 + laneMod;
                b[nt] = *(const v2f*)(bs + n * LDSS + ks + laneHalf * 2);
            }
#pragma unroll
            for (int mt = 0; mt < 2; ++mt)
#pragma unroll
                for (int nt = 0; nt < 2; ++nt)
                    c[mt][nt] = __builtin_amdgcn_wmma_f32_16x16x4_f32(
                        false, a[mt], false, b[nt], (short)0, c[mt][nt], false, false);
        }

        if (wave == 0) __builtin_amdgcn_s_wait_tensorcnt(0);
        __syncthreads();
        buf ^= 1;
    }

    // Write back C (VGPR r: M = r + 8*laneHalf; lanes carry N).
#pragma unroll
    for (int mt = 0; mt < 2; ++mt)
#pragma unroll
        for (int nt = 0; nt < 2; ++nt) {
            int n = nTile + wN + nt * 16 + laneMod;
#pragma unroll
            for (int r = 0; r < 8; ++r) {
                int m = mTile + wM + mt * 16 + laneHalf * 8 + r;
                Y[(size_t)m * N + n] = c[mt][nt][r];
            }
        }
}

// ---------------------------------------------------------------------------
// Fallback (workspace too small): direct per-batch-row rotation chain.
// ---------------------------------------------------------------------------
__global__ void kac_fallback(
    const float* __restrict__ x, const float* __restrict__ angles,
    const int* __restrict__ pi, const int* __restrict__ pj,
    const int* __restrict__ aidx, float* __restrict__ out,
    int dim, int ns, int batch)
{
    int b = blockIdx.x * blockDim.x + threadIdx.x;
    if (b >= batch) return;
    float*       row = out + (size_t)b * dim;
    const float* src = x   + (size_t)b * dim;
    for (int k = 0; k < dim; ++k) row[k] = src[k];
    for (int s = 0; s < ns; ++s) {
        int i = pi[s], j = pj[s];
        float sn, cs;
        sincosf(angles[aidx[s]], &sn, &cs);
        float xi = row[i], xj = row[j];
        row[i] = fmaf(cs, xi, sn * xj);
        row[j] = fmaf(-sn, xi, cs * xj);
    }
}

extern "C" void kernel_launch(void* const* d_in, const int* in_sizes, int n_in,
                              void* d_out, int out_size, void* d_ws, size_t ws_size,
                              hipStream_t stream)
{
    const float* x      = (const float*)d_in[0];
    const float* angles = (const float*)d_in[1];
    const int*   pi     = (const int*)d_in[2];
    const int*   pj     = (const int*)d_in[3];
    const int*   aidx   = (const int*)d_in[4];
    float*       out    = (float*)d_out;

    const int dim   = in_sizes[1];              // 4096
    const int ns    = in_sizes[2];              // 16384
    const int batch = in_sizes[0] / dim;        // 8192

    const size_t tableBytes = (size_t)ns * sizeof(float4);
    const size_t qBytes     = (size_t)dim * (size_t)dim * sizeof(float);

    const bool canTile = (dim % TILE_N == 0) && (batch % TILE_M == 0) && (dim % TILE_K == 0);

    if (canTile && ws_size >= tableBytes + qBytes) {
        float4* table = (float4*)d_ws;
        float*  q     = (float*)((char*)d_ws + tableBytes);

        kac_build_table<<<(ns + 255) / 256, 256, 0, stream>>>(angles, pi, pj, aidx, table, ns);

        size_t total = (size_t)dim * (size_t)dim;
        int initBlocks = (int)((total + 256 * 16 - 1) / (256 * 16));
        kac_init_q<<<initBlocks, 256, 0, stream>>>(q, dim);

        kac_rotate_q<<<(dim + 255) / 256, 256, 0, stream>>>(q, table, dim, ns);

        dim3 grid(dim / TILE_N, batch / TILE_M);
        kac_gemm_wmma<<<grid, 256, 0, stream>>>(x, q, out, batch, dim, dim);
    } else {
        kac_fallback<<<(batch + 255) / 256, 256, 0, stream>>>(
            x, angles, pi, pj, aidx, out, dim, ns, batch);
    }
}